// CTCLoss_23012434772416
// MI455X (gfx1250) — compile-verified
//
#include <hip/hip_runtime.h>
#include <hip/hip_bf16.h>
#include <cstdint>

// Problem shape (fixed by reference setup_inputs): B=32, T=1024, V=512, S=128
#define CTC_T 1024
#define CTC_V 512
#define CTC_S 128
#define CTC_L 257  // 2*S+1
#define NEGINF (-1e30f)

typedef __attribute__((ext_vector_type(2))) float v2f;
typedef __attribute__((ext_vector_type(4))) float v4f;
typedef __attribute__((ext_vector_type(8))) float v8f;

// b128 payload type the async-LDS builtin expects (per hipcc diagnostic:
// "int __vector(4) __device__ *" == int4 in address_space(1)).
typedef int v4i_t __attribute__((vector_size(16)));
typedef __attribute__((address_space(1))) v4i_t* g_v4i_ptr;
typedef __attribute__((address_space(3))) v4i_t* l_v4i_ptr;

// ---------------- CDNA5 async global->LDS path ----------------
#if defined(__gfx1250__) && __has_builtin(__builtin_amdgcn_global_load_async_to_lds_b128)
#define HAVE_ASYNC 1
#else
#define HAVE_ASYNC 0
#endif

__device__ __forceinline__ void async_copy16(const void* g, void* l) {
#if HAVE_ASYNC
  __builtin_amdgcn_global_load_async_to_lds_b128(
      (g_v4i_ptr)(uintptr_t)g,
      (l_v4i_ptr)(uint32_t)(uintptr_t)l,
      /*offset=*/0, /*cpol=*/0);
#else
  *(v4f*)l = *(const v4f*)g;  // synchronous fallback (compiler inserts waits)
#endif
}

#if HAVE_ASYNC
#if __has_builtin(__builtin_amdgcn_s_wait_asynccnt)
#define ASYNC_WAIT(n) __builtin_amdgcn_s_wait_asynccnt(n)
#else
#define ASYNC_WAIT(n) asm volatile("s_wait_asynccnt %0" ::"i"(n) : "memory")
#endif
#else
#define ASYNC_WAIT(n) ((void)0)
#endif

#if defined(__gfx1250__) && __has_builtin(__builtin_amdgcn_wmma_f32_16x16x4_f32)
#define HAVE_WMMA4 1
#else
#define HAVE_WMMA4 0
#endif

// =====================================================================
// Phase 1: logsumexp over V=512 for 16 consecutive time rows per block.
// One wave per block. 32 KB contiguous chunk -> LDS (async), row max via
// shuffles, Sum(exp) via v_wmma_f32_16x16x4_f32 against a ones B-matrix.
// =====================================================================
__global__ __launch_bounds__(32) void lse16_kernel(const float* __restrict__ logits,
                                                   float* __restrict__ lse) {
  constexpr int RS = CTC_V + 4;  // padded row stride in dwords (bank de-conflict)
  __shared__ __align__(16) float tile[16 * RS];  // 33024 B
  __shared__ float rowsum[16];

  const int lane = threadIdx.x;
  const float* src = logits + (size_t)blockIdx.x * (16 * CTC_V);

  // Stage 16 rows (contiguous 32 KB) into LDS with per-row padding.
#pragma unroll 4
  for (int i = 0; i < 64; ++i) {
    const int gb = i * 512 + lane * 16;  // byte offset inside chunk
    const int row = gb >> 11;            // /2048 bytes per row
    const int col = gb & 2047;
    async_copy16((const char*)src + gb, (char*)tile + row * (RS * 4) + col);
  }
  ASYNC_WAIT(0);
  __syncthreads();

  // Per-row max: 2 lanes per row (halves), shfl_xor(16) to combine.
  const int r = lane & 15;
  const int h = lane >> 4;
  float m = NEGINF;
  {
    const float* rp = &tile[r * RS + h * 256];
#pragma unroll 4
    for (int j = 0; j < 64; ++j) {
      v4f v = *(const v4f*)(rp + j * 4);
      m = fmaxf(m, fmaxf(fmaxf(v.x, v.y), fmaxf(v.z, v.w)));
    }
    m = fmaxf(m, __shfl_xor(m, 16, 32));
  }

#if HAVE_WMMA4
  // A-matrix 16x4 f32 layout: lanes 0-15 -> M=lane, K=0,1; lanes 16-31 -> M=lane-16, K=2,3.
  // B = ones(4x16) => D[m][n] = row-sum of A, replicated over n.
  v8f acc = {0.f, 0.f, 0.f, 0.f, 0.f, 0.f, 0.f, 0.f};
  v2f bones;
  bones.x = 1.0f;
  bones.y = 1.0f;
  const float* ap = &tile[r * RS + h * 2];
#pragma unroll 4
  for (int k = 0; k < 128; ++k) {
    v2f x = *(const v2f*)(ap + k * 4);
    v2f a;
    a.x = __expf(x.x - m);
    a.y = __expf(x.y - m);
    acc = __builtin_amdgcn_wmma_f32_16x16x4_f32(false, a, false, bones,
                                                (short)0, acc, false, false);
  }
  // D layout: VGPR q, lanes 0-15 -> M=q ; lanes 16-31 -> M=q+8 (replicated over N).
  if (lane == 0) {
#pragma unroll
    for (int q = 0; q < 8; ++q) rowsum[q] = acc[q];
  }
  if (lane == 16) {
#pragma unroll
    for (int q = 0; q < 8; ++q) rowsum[8 + q] = acc[q];
  }
  __syncthreads();
  if (lane < 16) lse[(size_t)blockIdx.x * 16 + lane] = m + __logf(rowsum[lane]);
#else
  float ssum = 0.0f;
  const float* rp2 = &tile[r * RS + h * 256];
  for (int j = 0; j < 256; ++j) ssum += __expf(rp2[j] - m);
  ssum += __shfl_xor(ssum, 16, 32);
  if (lane < 16) lse[(size_t)blockIdx.x * 16 + lane] = m + __logf(ssum);
#endif
}

// =====================================================================
// Phase 2: CTC forward DP. One block (288 threads = 9 waves) per batch
// element; lane s owns extended state s. Alpha double-buffered in LDS,
// one barrier per time step. Wave 0 async-prefetches the logits row
// PF=8 steps ahead into a 9-deep LDS ring (s_wait_asynccnt 4*(PF-1)
// retires exactly the oldest row each step; row index clamped so the
// in-flight count is constant — the wait amount must be an immediate).
// =====================================================================
__device__ __forceinline__ void issue_row(const float* g, float* l, int lane) {
#pragma unroll
  for (int i = 0; i < 4; ++i) {  // 4 x (32 lanes x 16B) = 2048 B
    const int off = i * 512 + lane * 16;
    async_copy16((const char*)g + off, (char*)l + off);
  }
}

__global__ __launch_bounds__(288) void ctc_dp_kernel(
    const float* __restrict__ logits, const int* __restrict__ targets,
    const int* __restrict__ loglen, const int* __restrict__ tgtlen,
    const float* __restrict__ lseg, float* __restrict__ out) {
  constexpr int PF = 8;
  constexpr int NBUF = PF + 1;

  __shared__ __align__(16) float rowbuf[NBUF][CTC_V];  // 18 KB ring
  __shared__ float abuf[2][CTC_L + 2];                 // alpha, 2 NEG pads in front
  __shared__ float lds_lse[CTC_T];                     // 4 KB

  const int b = blockIdx.x;
  const int tid = threadIdx.x;
  const int len = loglen[b];  // >= 1
  const int tl = tgtlen[b];
  const float* lg = logits + (size_t)b * CTC_T * CTC_V;

  for (int i = tid; i < CTC_T; i += 288) lds_lse[i] = lseg[b * CTC_T + i];
  if (tid < 2) {
    abuf[0][tid] = NEGINF;
    abuf[1][tid] = NEGINF;
  }

  // Per-lane extended label + skip-transition flag.
  const int s = tid;
  const bool act = (s < CTC_L);
  int ext = 0;  // blank
  bool skip = false;
  if (act && (s & 1)) {
    const int j = s >> 1;
    ext = targets[b * CTC_S + j];
    if (s >= 3) skip = (ext != targets[b * CTC_S + j - 1]);
  }

  // Prologue: rows 0..PF-1 in flight, wait for row 0.
  if (tid < 32) {
    for (int t = 0; t < PF; ++t) {
      const int rr = min(t, len - 1);
      issue_row(lg + (size_t)rr * CTC_V, rowbuf[t], tid);
    }
    ASYNC_WAIT(4 * (PF - 1));
  }
  __syncthreads();

  // t = 0 init: alpha0[0]=lp(blank), alpha0[1]=lp(label0) if tl>0, else NEG.
  float a = NEGINF;
  if (act) {
    const float lp = rowbuf[0][ext] - lds_lse[0];
    a = (s == 0 || (s == 1 && tl > 0)) ? lp : NEGINF;
    abuf[0][s + 2] = a;
  }
  if (tid < 32) {
    const int rr = min(PF, len - 1);
    issue_row(lg + (size_t)rr * CTC_V, rowbuf[PF % NBUF], tid);
    ASYNC_WAIT(4 * (PF - 1));
  }
  __syncthreads();

  int cur = 0;
  int bufc = 0;  // ring slot holding row t (t=0 so far)
  for (int t = 1; t < len; ++t) {
    const int pbuf = bufc;  // slot of row t-1, free now; (t+PF) % NBUF == (t-1) % NBUF
    bufc = (bufc + 1 == NBUF) ? 0 : bufc + 1;
    const int nxt = cur ^ 1;
    if (act) {
      const float lp = rowbuf[bufc][ext] - lds_lse[t];
      const float a1 = abuf[cur][s + 1];                 // alpha[s-1]
      const float a2 = skip ? abuf[cur][s] : NEGINF;     // alpha[s-2]
      const float z = fmaxf(a, fmaxf(a1, a2));
      const float e = __expf(a - z) + __expf(a1 - z) + __expf(a2 - z);
      a = z + __logf(e) + lp;
      abuf[nxt][s + 2] = a;
    }
    if (tid < 32) {
      const int rr = min(t + PF, len - 1);  // clamp: keeps PF rows in flight
      issue_row(lg + (size_t)rr * CTC_V, rowbuf[pbuf], tid);
      ASYNC_WAIT(4 * (PF - 1));  // oldest outstanding row (t+1) done
    }
    __syncthreads();
    cur = nxt;
  }

  if (tid == 0) {
    ASYNC_WAIT(0);
    const float ab = abuf[cur][2 * tl + 2];
    const float al = (tl > 0) ? abuf[cur][max(2 * tl - 1, 0) + 2] : NEGINF;
    const float z = fmaxf(ab, al);
    out[b] = -(z + __logf(__expf(ab - z) + __expf(al - z)));
  }
}

// =====================================================================
extern "C" void kernel_launch(void* const* d_in, const int* in_sizes, int n_in,
                              void* d_out, int out_size, void* d_ws, size_t ws_size,
                              hipStream_t stream) {
  const float* logits = (const float*)d_in[0];
  const int* targets = (const int*)d_in[1];
  const int* loglen = (const int*)d_in[2];
  const int* tgtlen = (const int*)d_in[3];
  float* out = (float*)d_out;
  float* lse = (float*)d_ws;  // B*T floats = 128 KB of workspace

  const int B = in_sizes[2];  // logits_lengths has B elements
  const int nchunks = (B * CTC_T) / 16;

  lse16_kernel<<<dim3(nchunks), dim3(32), 0, stream>>>(logits, lse);
  ctc_dp_kernel<<<dim3(B), dim3(288), 0, stream>>>(logits, targets, loglen,
                                                   tgtlen, lse, out);
}